// RWKV_LAYER_73615739453479
// MI455X (gfx1250) — compile-verified
//
#include <hip/hip_runtime.h>
#include <cstddef>
#include <cstdint>

static constexpr int DIM = 2048;
static constexpr int HID = 8192;
static constexpr int NLAYER = 4;

// WMMA GEMV staging geometry
static constexpr int KC      = 32;              // k-chunk per staging step
static constexpr int WSTRIDE = 36;              // padded LDS row stride (floats), 144B = 9*16B
static constexpr int CHUNK_F = 16 * WSTRIDE;    // floats per staging buffer (576)

typedef float v2f __attribute__((ext_vector_type(2)));
typedef float v8f __attribute__((ext_vector_type(8)));

__device__ __forceinline__ float sigmoidf_(float x) { return 1.0f / (1.0f + expf(-x)); }

// ---------------------------------------------------------------------------
// Async-stage one 16-row x KC chunk of row-major W (row stride K floats) into
// this wave's LDS buffer using the CDNA5 async copy engine (ASYNCcnt).
// 4 instructions x (32 lanes x 16B) = 2KB, fully coalesced along rows.
// ---------------------------------------------------------------------------
__device__ __forceinline__ void issue_chunk_async(const float* __restrict__ g,
                                                  unsigned ldsb, int K)
{
    const int lane = threadIdx.x & 31;
    const float* gp = g + (size_t)(lane >> 3) * (size_t)K + (lane & 7) * 4;
    const unsigned lp = ldsb + (unsigned)(((lane >> 3) * WSTRIDE + (lane & 7) * 4) * 4);
#pragma unroll
    for (int i = 0; i < 4; ++i) {
        asm volatile("global_load_async_to_lds_b128 %0, %1, off"
                     :: "v"(lp + (unsigned)(i * 4 * WSTRIDE * 4)),
                        "v"(gp + (size_t)i * 4 * (size_t)K)
                     : "memory");
    }
}

// ---------------------------------------------------------------------------
// WMMA over one staged chunk. A tile (16x4 f32): lanes 0-15 -> M=lane,
// K={k,k+1}; lanes 16-31 -> M=lane-16, K={k+2,k+3}. B = x broadcast columns.
// LDS reads are bank-conflict-free (36*lane mod 64 distinct, halves disjoint).
// ---------------------------------------------------------------------------
__device__ __forceinline__ v8f wmma_chunk(const float* __restrict__ wb,
                                          const float* __restrict__ xs, v8f acc)
{
    const int lane = threadIdx.x & 31;
    const int kh   = (lane >> 4) << 1;
    const float* wr = wb + (lane & 15) * WSTRIDE + kh;
#pragma unroll
    for (int kc = 0; kc < KC; kc += 4) {
        v2f a, b;
        a.x = wr[kc];
        a.y = wr[kc + 1];
        b.x = xs[kc + kh];
        b.y = xs[kc + kh + 1];
        acc = __builtin_amdgcn_wmma_f32_16x16x4_f32(false, a, false, b, (short)0, acc,
                                                    false, false);
    }
    return acc;
}

// ---------------------------------------------------------------------------
// Full GEMV for 16 rows [row0,row0+16): async double-buffered weight staging,
// WMMA accumulate. wb = this wave's private 2*CHUNK_F float LDS region.
// ---------------------------------------------------------------------------
__device__ __forceinline__ v8f wmma_gemv_async(const float* __restrict__ W, int K,
                                               int row0, const float* __restrict__ xs,
                                               float* __restrict__ wb, v8f acc)
{
    const unsigned wb_byte = (unsigned)(uintptr_t)wb;   // low 32 bits = LDS offset
    const float* wrow = W + (size_t)row0 * (size_t)K;
    const int nch = K / KC;
    issue_chunk_async(wrow, wb_byte, K);
    for (int c = 0; c < nch; ++c) {
        if (c + 1 < nch) {
            issue_chunk_async(wrow + (size_t)(c + 1) * KC,
                              wb_byte + (unsigned)(((c + 1) & 1) * CHUNK_F * 4), K);
            asm volatile("s_wait_asynccnt 0x4" ::: "memory");  // chunk c landed
        } else {
            asm volatile("s_wait_asynccnt 0x0" ::: "memory");
        }
        acc = wmma_chunk(wb + (c & 1) * CHUNK_F, xs + c * KC, acc);
    }
    return acc;
}

// ---------------------------------------------------------------------------
// rrk = key[l] @ xn + keymul[l] @ state0     (3*DIM outputs, K = DIM each)
// ---------------------------------------------------------------------------
__global__ void k_gemv_rrk(const float* __restrict__ keyW,
                           const float* __restrict__ kmulW,
                           const float* __restrict__ xn,
                           const float* __restrict__ s0,
                           float* __restrict__ rrk)
{
    __shared__ __align__(16) float xs[2 * DIM];
    __shared__ __align__(16) float wbuf[8][2 * CHUNK_F];
    for (int i = threadIdx.x; i < DIM; i += blockDim.x) {
        xs[i]       = xn[i];
        xs[DIM + i] = s0[i];
    }
    __syncthreads();
    const int wave = threadIdx.x >> 5;
    const int row0 = (blockIdx.x * 8 + wave) * 16;
    v8f acc = {};
    acc = wmma_gemv_async(keyW,  DIM, row0, xs,       wbuf[wave], acc);
    acc = wmma_gemv_async(kmulW, DIM, row0, xs + DIM, wbuf[wave], acc);
    const int lane = threadIdx.x & 31;
    if ((lane & 15) == 0) {
        const int base = row0 + ((lane >> 4) << 3);
#pragma unroll
        for (int r = 0; r < 8; ++r) rrk[base + r] = acc[r];
    }
}

// sx = xcur + outputv[l] @ t
__global__ void k_gemv_out(const float* __restrict__ W, const float* __restrict__ x,
                           const float* __restrict__ xc, float* __restrict__ sx)
{
    __shared__ __align__(16) float xs[DIM];
    __shared__ __align__(16) float wbuf[8][2 * CHUNK_F];
    for (int i = threadIdx.x; i < DIM; i += blockDim.x) xs[i] = x[i];
    __syncthreads();
    const int wave = threadIdx.x >> 5;
    const int row0 = (blockIdx.x * 8 + wave) * 16;
    v8f acc = {};
    acc = wmma_gemv_async(W, DIM, row0, xs, wbuf[wave], acc);
    const int lane = threadIdx.x & 31;
    if ((lane & 15) == 0) {
        const int base = row0 + ((lane >> 4) << 3);
#pragma unroll
        for (int r = 0; r < 8; ++r) sx[base + r] = xc[base + r] + acc[r];
    }
}

// kk = relu(key_ffn[l] @ dx)^2            (HID outputs, K = DIM)
__global__ void k_gemv_kffn(const float* __restrict__ W, const float* __restrict__ x,
                            float* __restrict__ kk)
{
    __shared__ __align__(16) float xs[DIM];
    __shared__ __align__(16) float wbuf[8][2 * CHUNK_F];
    for (int i = threadIdx.x; i < DIM; i += blockDim.x) xs[i] = x[i];
    __syncthreads();
    const int wave = threadIdx.x >> 5;
    const int row0 = (blockIdx.x * 8 + wave) * 16;
    v8f acc = {};
    acc = wmma_gemv_async(W, DIM, row0, xs, wbuf[wave], acc);
    const int lane = threadIdx.x & 31;
    if ((lane & 15) == 0) {
        const int base = row0 + ((lane >> 4) << 3);
#pragma unroll
        for (int r = 0; r < 8; ++r) {
            float v = fmaxf(acc[r], 0.0f);
            kk[base + r] = v * v;
        }
    }
}

// r = sigmoid(rffn[l] @ xr)
__global__ void k_gemv_rffn(const float* __restrict__ W, const float* __restrict__ x,
                            float* __restrict__ rv)
{
    __shared__ __align__(16) float xs[DIM];
    __shared__ __align__(16) float wbuf[8][2 * CHUNK_F];
    for (int i = threadIdx.x; i < DIM; i += blockDim.x) xs[i] = x[i];
    __syncthreads();
    const int wave = threadIdx.x >> 5;
    const int row0 = (blockIdx.x * 8 + wave) * 16;
    v8f acc = {};
    acc = wmma_gemv_async(W, DIM, row0, xs, wbuf[wave], acc);
    const int lane = threadIdx.x & 31;
    if ((lane & 15) == 0) {
        const int base = row0 + ((lane >> 4) << 3);
#pragma unroll
        for (int r = 0; r < 8; ++r) rv[base + r] = sigmoidf_(acc[r]);
    }
}

// xnew = sx + r * (vffn[l] @ kk)          (DIM outputs, K = HID); 4 waves/block
__global__ void k_gemv_vffn(const float* __restrict__ W, const float* __restrict__ x,
                            const float* __restrict__ rv, const float* __restrict__ sx,
                            float* __restrict__ xnew)
{
    __shared__ __align__(16) float xs[HID];
    __shared__ __align__(16) float wbuf[4][2 * CHUNK_F];
    for (int i = threadIdx.x; i < HID; i += blockDim.x) xs[i] = x[i];
    __syncthreads();
    const int wave = threadIdx.x >> 5;
    const int row0 = (blockIdx.x * 4 + wave) * 16;
    v8f acc = {};
    acc = wmma_gemv_async(W, HID, row0, xs, wbuf[wave], acc);
    const int lane = threadIdx.x & 31;
    if ((lane & 15) == 0) {
        const int base = row0 + ((lane >> 4) << 3);
#pragma unroll
        for (int r = 0; r < 8; ++r) xnew[base + r] = sx[base + r] + rv[base + r] * acc[r];
    }
}

// ---------------------------------------------------------------------------
// LayerNorm 1: xn = LN(x)*w+b ; write to ws.xn and state_out row (5l+0)
// ---------------------------------------------------------------------------
__global__ void k_ln1(const float* __restrict__ x, const float* __restrict__ w,
                      const float* __restrict__ b, float* __restrict__ xn,
                      float* __restrict__ st_xn)
{
    __shared__ float ssum[256], ssq[256];
    const int tid = threadIdx.x;
    float s = 0.0f, q = 0.0f;
    for (int i = tid; i < DIM; i += 256) { float v = x[i]; s += v; q += v * v; }
    ssum[tid] = s; ssq[tid] = q;
    __syncthreads();
    for (int off = 128; off > 0; off >>= 1) {
        if (tid < off) { ssum[tid] += ssum[tid + off]; ssq[tid] += ssq[tid + off]; }
        __syncthreads();
    }
    const float mean = ssum[0] * (1.0f / DIM);
    const float var  = ssq[0] * (1.0f / DIM) - mean * mean;
    const float rstd = rsqrtf(var + 1e-5f);
    for (int i = tid; i < DIM; i += 256) {
        float v = (x[i] - mean) * rstd * w[i] + b[i];
        xn[i] = v;
        st_xn[i] = v;
    }
}

// LayerNorm 2 + token-shift mixes
__global__ void k_ln2(const float* __restrict__ sx, const float* __restrict__ w,
                      const float* __restrict__ b, const float* __restrict__ sffn,
                      const float* __restrict__ tmk, const float* __restrict__ tmr,
                      float* __restrict__ st_x2n, float* __restrict__ dx,
                      float* __restrict__ xr)
{
    __shared__ float ssum[256], ssq[256];
    const int tid = threadIdx.x;
    float s = 0.0f, q = 0.0f;
    for (int i = tid; i < DIM; i += 256) { float v = sx[i]; s += v; q += v * v; }
    ssum[tid] = s; ssq[tid] = q;
    __syncthreads();
    for (int off = 128; off > 0; off >>= 1) {
        if (tid < off) { ssum[tid] += ssum[tid + off]; ssq[tid] += ssq[tid + off]; }
        __syncthreads();
    }
    const float mean = ssum[0] * (1.0f / DIM);
    const float var  = ssq[0] * (1.0f / DIM) - mean * mean;
    const float rstd = rsqrtf(var + 1e-5f);
    for (int i = tid; i < DIM; i += 256) {
        float x2n = (sx[i] - mean) * rstd * w[i] + b[i];
        float sf  = sffn[i];
        st_x2n[i] = x2n;
        dx[i] = x2n + sf * tmk[i];
        xr[i] = x2n + sf * tmr[i];
    }
}

// ---------------------------------------------------------------------------
// WKV elementwise recurrence
// ---------------------------------------------------------------------------
__global__ void k_wkv(const float* __restrict__ rrk,
                      const float* __restrict__ aa, const float* __restrict__ bb,
                      const float* __restrict__ pp, const float* __restrict__ tf,
                      const float* __restrict__ td, float* __restrict__ t,
                      float* __restrict__ naa, float* __restrict__ nbb,
                      float* __restrict__ npp)
{
    const int i = blockIdx.x * blockDim.x + threadIdx.x;
    if (i >= DIM) return;
    const float k  = rrk[i];
    const float v  = rrk[DIM + i];
    const float rr = rrk[2 * DIM + i];
    const float rsig = sigmoidf_(rr);
    const float a0 = aa[i], b0 = bb[i], p0 = pp[i];

    const float ww = tf[i] + k;
    const float qq = fmaxf(p0, ww);
    const float e1 = expf(p0 - qq);
    const float e2 = expf(ww - qq);
    const float an = e1 * a0 + e2 * v;
    const float bn = e1 * b0 + e2;

    const float ww2 = p0 + td[i];
    const float q2  = fmaxf(ww2, k);
    const float f1  = expf(ww2 - q2);
    const float f2  = expf(k - q2);

    t[i]   = rsig * (an / bn);
    naa[i] = f1 * a0 + f2 * v;
    nbb[i] = f1 * b0 + f2;
    npp[i] = q2;
}

// ---------------------------------------------------------------------------
extern "C" void kernel_launch(void* const* d_in, const int* in_sizes, int n_in,
                              void* d_out, int out_size, void* d_ws, size_t ws_size,
                              hipStream_t stream)
{
    (void)in_sizes; (void)n_in; (void)out_size; (void)ws_size;
    const float* x_in  = (const float*)d_in[0];
    const float* state = (const float*)d_in[1];
    const float* ln1w  = (const float*)d_in[2];
    const float* ln1b  = (const float*)d_in[3];
    const float* ln2w  = (const float*)d_in[4];
    const float* ln2b  = (const float*)d_in[5];
    const float* td    = (const float*)d_in[6];
    const float* tf    = (const float*)d_in[7];
    const float* keyW  = (const float*)d_in[8];
    const float* kmulW = (const float*)d_in[9];
    const float* outvW = (const float*)d_in[10];
    const float* tmk   = (const float*)d_in[11];
    const float* tmr   = (const float*)d_in[12];
    const float* kffnW = (const float*)d_in[13];
    const float* rffnW = (const float*)d_in[14];
    const float* vffnW = (const float*)d_in[15];

    float* out       = (float*)d_out;          // [DIM] x_out
    float* state_out = out + DIM;              // [5*NLAYER, DIM]

    float* ws   = (float*)d_ws;
    float* xcur = ws;                // DIM
    float* xn   = ws + DIM;          // DIM
    float* rrk  = ws + 2 * DIM;      // 3*DIM
    float* tbuf = ws + 5 * DIM;      // DIM
    float* sx   = ws + 6 * DIM;      // DIM
    float* dx   = ws + 7 * DIM;      // DIM
    float* xr   = ws + 8 * DIM;      // DIM
    float* rv   = ws + 9 * DIM;      // DIM
    float* kk   = ws + 10 * DIM;     // HID

    hipMemcpyAsync(xcur, x_in, DIM * sizeof(float), hipMemcpyDeviceToDevice, stream);

    const size_t DD = (size_t)DIM * DIM;
    for (int l = 0; l < NLAYER; ++l) {
        const float* st  = state + (size_t)(5 * l) * DIM;      // input state rows
        float*       sto = state_out + (size_t)(5 * l) * DIM;  // output state rows

        k_ln1<<<1, 256, 0, stream>>>(xcur, ln1w + (size_t)l * DIM, ln1b + (size_t)l * DIM,
                                     xn, sto);
        k_gemv_rrk<<<(3 * DIM / 16) / 8, 256, 0, stream>>>(
            keyW + (size_t)l * 3 * DD, kmulW + (size_t)l * 3 * DD, xn, st, rrk);
        k_wkv<<<DIM / 256, 256, 0, stream>>>(
            rrk, st + DIM, st + 2 * DIM, st + 3 * DIM,
            tf + (size_t)l * DIM, td + (size_t)l * DIM, tbuf,
            sto + DIM, sto + 2 * DIM, sto + 3 * DIM);
        k_gemv_out<<<(DIM / 16) / 8, 256, 0, stream>>>(outvW + (size_t)l * DD, tbuf, xcur, sx);
        k_ln2<<<1, 256, 0, stream>>>(sx, ln2w + (size_t)l * DIM, ln2b + (size_t)l * DIM,
                                     st + 4 * DIM, tmk + (size_t)l * DIM,
                                     tmr + (size_t)l * DIM, sto + 4 * DIM, dx, xr);
        k_gemv_kffn<<<(HID / 16) / 8, 256, 0, stream>>>(kffnW + (size_t)l * HID * DIM, dx, kk);
        k_gemv_rffn<<<(DIM / 16) / 8, 256, 0, stream>>>(rffnW + (size_t)l * DD, xr, rv);
        k_gemv_vffn<<<(DIM / 16) / 4, 128, 0, stream>>>(vffnW + (size_t)l * DIM * HID, kk, rv,
                                                        sx, xcur);
    }
    hipMemcpyAsync(out, xcur, DIM * sizeof(float), hipMemcpyDeviceToDevice, stream);
}